// FIPE_79817672229536
// MI455X (gfx1250) — compile-verified
//
#include <hip/hip_runtime.h>

typedef float v2f __attribute__((ext_vector_type(2)));
typedef float v4f __attribute__((ext_vector_type(4)));
typedef float v8f __attribute__((ext_vector_type(8)));

#define IMG_H 512
#define IMG_W 512
#define N_IMG 256                 // 8 * 32
#define TILES_X 32                // 512 / 16
#define TILES_PER_IMG 1024        // 32 * 32
#define TOTAL_TILES 262144        // 256 * 1024
#define TPW 4                     // tiles per wave
#define WAVES_PER_BLOCK 8
#define LDS_STRIDE 20             // 16 cols + pad, keeps float4 16B-aligned

// D = sum_k A_slice[k] * B_slice[k] + C  over the full K=16 via 4x K=4 WMMAs
__device__ __forceinline__ v8f wmma16(const v2f a[4], const v2f b[4], v8f c) {
#pragma unroll
    for (int k = 0; k < 4; ++k) {
        c = __builtin_amdgcn_wmma_f32_16x16x4_f32(
                /*neg_a=*/false, a[k], /*neg_b=*/false, b[k],
                /*c_mod=*/(short)0, c, /*reuse_a=*/false, /*reuse_b=*/false);
    }
    return c;
}

// B-fragment (4x16 slices of a 16x16 matrix M) from LDS: lane n=L&15 holds
// column n; lane half selects K pair.  b[k0] elem e <- M[k0*4 + 2*(L>>4) + e][n]
__device__ __forceinline__ void load_bfrag(const float* lds, int lane, v2f b[4]) {
    const int n  = lane & 15;
    const int kb = (lane >> 4) * 2;
#pragma unroll
    for (int k0 = 0; k0 < 4; ++k0) {
        const int kk = k0 * 4 + kb;
        b[k0].x = lds[kk * LDS_STRIDE + n];
        b[k0].y = lds[(kk + 1) * LDS_STRIDE + n];
    }
}

// Store D-fragment transposed: lds[col][row] = D[row][col]  (lds <- D^T)
__device__ __forceinline__ void store_d_T(float* lds, int lane, v8f d) {
    const int col   = lane & 15;
    const int rbase = (lane >> 4) * 8;
#pragma unroll
    for (int r = 0; r < 8; ++r)
        lds[col * LDS_STRIDE + rbase + r] = d[r];
}

__global__ __launch_bounds__(256) void dct_lowpass_wmma(
        const float* __restrict__ x, const float* __restrict__ A,
        const float* __restrict__ mask,
        float* __restrict__ out_low, float* __restrict__ out_high) {
    __shared__ __align__(16) float smem[WAVES_PER_BLOCK * 2 * 16 * LDS_STRIDE];

    const int lane = threadIdx.x & 31;
    const int wave = threadIdx.x >> 5;
    float* ldsX = smem + wave * (2 * 16 * LDS_STRIDE); // original tile
    float* ldsT = ldsX + 16 * LDS_STRIDE;              // stage scratch

    // ---- per-wave constants --------------------------------------------
    // A-fragments for Bd = diag(A,A) and BdT = diag(A^T,A^T).
    // Lane L holds row m=L&15, K pair kb=2*(L>>4): elem e <- op[m][k0*4+kb+e]
    const int m  = lane & 15;
    const int kb = (lane >> 4) * 2;
    v2f a_fwd[4], a_bwd[4];
#pragma unroll
    for (int k0 = 0; k0 < 4; ++k0) {
#pragma unroll
        for (int e = 0; e < 2; ++e) {
            const int k = k0 * 4 + kb + e;
            const bool same_half = ((m >> 3) == (k >> 3));
            const float f = same_half ? A[(m & 7) * 8 + (k & 7)] : 0.0f;
            const float g = same_half ? A[(k & 7) * 8 + (m & 7)] : 0.0f;
            a_fwd[k0][e] = f;   // Bd[m][k]
            a_bwd[k0][e] = g;   // BdT[m][k]
        }
    }
    // mask factor for F^T element (row=r+8*(L>>4), col=L&15): mask[L&7][r]
    float mrow[8];
#pragma unroll
    for (int r = 0; r < 8; ++r) mrow[r] = mask[(lane & 7) * 8 + r];

    // ---- tile loop ------------------------------------------------------
    const int gwave = blockIdx.x * WAVES_PER_BLOCK + wave;
#pragma unroll 1
    for (int i = 0; i < TPW; ++i) {
        const int tile = gwave * TPW + i;
        if (tile >= TOTAL_TILES) break;               // wave-uniform
        const int img = tile >> 10;
        const int wi  = tile & (TILES_PER_IMG - 1);
        const int h0  = (wi >> 5) << 4;
        const int w0  = (wi & (TILES_X - 1)) << 4;
        const size_t ibase = (size_t)img * (IMG_H * IMG_W);
        const float* xg = x + ibase;

        // load 16x16 tile into ldsX with coalesced float4 (2 per lane)
#pragma unroll
        for (int t = 0; t < 2; ++t) {
            const int chunk = lane + 32 * t;          // 0..63
            const int row = chunk >> 2;
            const int c4  = (chunk & 3) * 4;
            v4f v = *(const v4f*)(xg + (size_t)(h0 + row) * IMG_W + w0 + c4);
            *(v4f*)(ldsX + row * LDS_STRIDE + c4) = v;
            if (i + 1 < TPW)                          // warm next tile (L2)
                __builtin_prefetch(xg + (size_t)(h0 + row) * IMG_W + w0 + 16 + c4, 0, 1);
        }

        v2f b[4];
        v8f acc;

        // Stage 1: U = Bd * X           -> ldsT = U^T
        load_bfrag(ldsX, lane, b);
        acc = v8f{};
        acc = wmma16(a_fwd, b, acc);
        store_d_T(ldsT, lane, acc);

        // Stage 2: V = Bd * U^T = F^T ; mask ; -> ldsT = F_m
        load_bfrag(ldsT, lane, b);
        acc = v8f{};
        acc = wmma16(a_fwd, b, acc);
#pragma unroll
        for (int r = 0; r < 8; ++r) acc[r] *= mrow[r];
        store_d_T(ldsT, lane, acc);

        // Stage 3: W = BdT * F_m        -> ldsT = W^T
        load_bfrag(ldsT, lane, b);
        acc = v8f{};
        acc = wmma16(a_bwd, b, acc);
        store_d_T(ldsT, lane, acc);

        // Stage 4: Z = BdT * W^T = x_low^T -> ldsT = x_low
        load_bfrag(ldsT, lane, b);
        acc = v8f{};
        acc = wmma16(a_bwd, b, acc);
        store_d_T(ldsT, lane, acc);

        // write x_low and x_high = x - x_low, coalesced float4
#pragma unroll
        for (int t = 0; t < 2; ++t) {
            const int chunk = lane + 32 * t;
            const int row = chunk >> 2;
            const int c4  = (chunk & 3) * 4;
            v4f xv = *(const v4f*)(ldsX + row * LDS_STRIDE + c4);
            v4f lv = *(const v4f*)(ldsT + row * LDS_STRIDE + c4);
            const size_t off = ibase + (size_t)(h0 + row) * IMG_W + w0 + c4;
            *(v4f*)(out_low + off)  = lv;
            *(v4f*)(out_high + off) = xv - lv;
        }
    }
}

extern "C" void kernel_launch(void* const* d_in, const int* in_sizes, int n_in,
                              void* d_out, int out_size, void* d_ws, size_t ws_size,
                              hipStream_t stream) {
    const float* x    = (const float*)d_in[0];
    const float* A    = (const float*)d_in[1];
    const float* mask = (const float*)d_in[2];
    float* out_low  = (float*)d_out;
    float* out_high = out_low + (size_t)N_IMG * IMG_H * IMG_W;

    const int blocks = TOTAL_TILES / (TPW * WAVES_PER_BLOCK); // 8192
    dct_lowpass_wmma<<<blocks, 256, 0, stream>>>(x, A, mask, out_low, out_high);
}